// MyGATCN_75385265979975
// MI455X (gfx1250) — compile-verified
//
#include <hip/hip_runtime.h>
#include <math.h>

// Problem constants (match reference)
#define NNODES 10000
#define FIN    512
#define HEADS  8
#define C1V    30
#define CHV    64
#define NCLSV  64
#define NEDGE  160000
#define EEDGE  (NEDGE + NNODES)   // edges + self loops = 170000
#define HC1    (HEADS * C1V)      // 240
#define HC2    (HEADS * CHV)      // 512

typedef __attribute__((ext_vector_type(2))) float v2f;
typedef __attribute__((ext_vector_type(8))) float v8f;

// ---------------------------------------------------------------------------
// fp32 WMMA GEMM: C[M,N] = A[M,K] * B[K,N] (+bias[N]), row-major everything.
// One wave computes a 16 x (16*NT) output strip: NT accumulators share one
// A fragment per K-step (v_wmma_f32_16x16x4_f32), amortizing A traffic NT x.
// M%16==0, (N % (16*NT))==0, K%4==0 guaranteed by caller.
// ---------------------------------------------------------------------------
template <int NT>
__global__ __launch_bounds__(32)
void k_gemm_wmma_f32(const float* __restrict__ A, const float* __restrict__ B,
                     const float* __restrict__ bias, float* __restrict__ C,
                     int M, int K, int N)
{
    const int lane = threadIdx.x & 31;
    const int half = lane >> 4;       // 0: lanes 0-15, 1: lanes 16-31
    const int li   = lane & 15;
    const int row0 = blockIdx.x * 16;
    const int col0 = blockIdx.y * (16 * NT);

    const float* Arow = A + (size_t)(row0 + li) * K + 2 * half; // K offset base
    const float* Bp   = B + (size_t)(2 * half) * N + col0 + li;

    v8f acc[NT];
#pragma unroll
    for (int t = 0; t < NT; ++t) acc[t] = (v8f){};

    for (int k = 0; k < K; k += 4) {
        // A frag: reg r holds A[row0+li][K = k + 2*half + r] (contiguous pair)
        v2f a = *(const v2f*)(Arow + k);
        const float* Bk = Bp + (size_t)k * N;
#pragma unroll
        for (int t = 0; t < NT; ++t) {
            // B frag: reg r holds B[K = k + 2*half + r][col0 + 16*t + li]
            v2f b;
            b[0] = Bk[16 * t];
            b[1] = Bk[16 * t + (size_t)N];
            acc[t] = __builtin_amdgcn_wmma_f32_16x16x4_f32(
                /*neg_a=*/false, a, /*neg_b=*/false, b,
                /*c_mod=*/(short)0, acc[t], /*reuse_a=*/false, /*reuse_b=*/false);
        }
    }

#pragma unroll
    for (int t = 0; t < NT; ++t) {
        const int col = col0 + 16 * t + li;
        const float bv = bias ? bias[col] : 0.0f;
#pragma unroll
        for (int r = 0; r < 8; ++r) {
            // C/D layout: reg r -> M = r + 8*half, N = li
            C[(size_t)(row0 + 8 * half + r) * N + col] = acc[t][r] + bv;
        }
    }
}

// ---------------------------------------------------------------------------
// Per-node attention scores: asrc[n,h] = sum_c t[n,h,c]*a_src[h,c] (adst alike)
// ---------------------------------------------------------------------------
__global__ __launch_bounds__(256)
void k_attn_scores(const float* __restrict__ t, const float* __restrict__ a_src,
                   const float* __restrict__ a_dst, float* __restrict__ asrc,
                   float* __restrict__ adst, int C)
{
    int tid = blockIdx.x * blockDim.x + threadIdx.x;   // n*HEADS + h
    if (tid >= NNODES * HEADS) return;
    const int h = tid % HEADS;
    const int n = tid / HEADS;
    const float* tp = t + (size_t)n * (HEADS * C) + h * C;
    const float* as = a_src + h * C;
    const float* ad = a_dst + h * C;
    float s1 = 0.0f, s2 = 0.0f;
    for (int c = 0; c < C; ++c) {
        const float v = tp[c];
        s1 += v * as[c];
        s2 += v * ad[c];
    }
    asrc[tid] = s1;
    adst[tid] = s2;
}

__device__ __forceinline__ void edge_nodes(const int* __restrict__ ei, int e,
                                           int& s, int& d)
{
    if (e < NEDGE) { s = ei[e]; d = ei[NEDGE + e]; }
    else           { s = d = e - NEDGE; }           // appended self loop
}

// ordered-uint mapping: monotone in float value
__device__ __forceinline__ unsigned f2ord(float f) {
    unsigned u = __float_as_uint(f);
    return (u & 0x80000000u) ? ~u : (u | 0x80000000u);
}
__device__ __forceinline__ float ord2f(unsigned u) {
    return (u & 0x80000000u) ? __uint_as_float(u ^ 0x80000000u)
                             : __uint_as_float(~u);
}

// ---------------------------------------------------------------------------
// Phase 1: alpha = leaky_relu(asrc[src]+adst[dst], 0.2); segment max into mu
// ---------------------------------------------------------------------------
__global__ __launch_bounds__(256)
void k_edge_alpha_max(const int* __restrict__ ei, const float* __restrict__ asrc,
                      const float* __restrict__ adst, float* __restrict__ alpha,
                      unsigned* __restrict__ mu)
{
    int tid = blockIdx.x * blockDim.x + threadIdx.x;   // e*HEADS + h
    if (tid >= EEDGE * HEADS) return;
    const int h = tid % HEADS;
    const int e = tid / HEADS;
    int s, d; edge_nodes(ei, e, s, d);
    float a = asrc[s * HEADS + h] + adst[d * HEADS + h];
    a = a > 0.0f ? a : 0.2f * a;
    alpha[tid] = a;
    atomicMax(&mu[d * HEADS + h], f2ord(a));
}

// ---------------------------------------------------------------------------
// Phase 2: e = exp(alpha - m[dst]); segment sum into ssum; alpha <- e
// ---------------------------------------------------------------------------
__global__ __launch_bounds__(256)
void k_edge_expsum(const int* __restrict__ ei, const unsigned* __restrict__ mu,
                   float* __restrict__ alpha, float* __restrict__ ssum)
{
    int tid = blockIdx.x * blockDim.x + threadIdx.x;
    if (tid >= EEDGE * HEADS) return;
    const int h = tid % HEADS;
    const int e = tid / HEADS;
    int s, d; edge_nodes(ei, e, s, d);
    const float m  = ord2f(mu[d * HEADS + h]);
    const float ev = expf(alpha[tid] - m);
    alpha[tid] = ev;
    atomicAdd(&ssum[d * HEADS + h], ev);
}

// ---------------------------------------------------------------------------
// Phase 2b: normalize in place: alpha[e,h] <- alpha[e,h] / (ssum[dst,h]+eps)
// (weight varies only per edge-head; avoids divide + ssum load per feature)
// ---------------------------------------------------------------------------
__global__ __launch_bounds__(256)
void k_edge_norm(const int* __restrict__ ei, const float* __restrict__ ssum,
                 float* __restrict__ alpha)
{
    int tid = blockIdx.x * blockDim.x + threadIdx.x;
    if (tid >= EEDGE * HEADS) return;
    const int h = tid % HEADS;
    const int e = tid / HEADS;
    int s, d; edge_nodes(ei, e, s, d);
    alpha[tid] = alpha[tid] / (ssum[d * HEADS + h] + 1e-16f);
}

// ---------------------------------------------------------------------------
// Phase 3: agg[dst, f] += w[e,h] * t[src, f], f in [0, HEADS*C)
// ---------------------------------------------------------------------------
__global__ __launch_bounds__(256)
void k_edge_aggregate(const int* __restrict__ ei, const float* __restrict__ w,
                      const float* __restrict__ t, float* __restrict__ agg, int C)
{
    const int HC = HEADS * C;
    int tid = blockIdx.x * blockDim.x + threadIdx.x;   // e*HC + f
    if (tid >= EEDGE * HC) return;
    const int f = tid % HC;
    const int e = tid / HC;
    const int h = f / C;
    int s, d; edge_nodes(ei, e, s, d);
    const float we = w[(size_t)e * HEADS + h];
    atomicAdd(&agg[(size_t)d * HC + f], we * t[(size_t)s * HC + f]);
}

// ---------------------------------------------------------------------------
// out[n,f] = in[n,f] + bias[f]; optional ELU (alpha=1, expm1)
// ---------------------------------------------------------------------------
__global__ __launch_bounds__(256)
void k_bias_act(const float* __restrict__ in, const float* __restrict__ bias,
                float* __restrict__ out, int F, int do_elu)
{
    int tid = blockIdx.x * blockDim.x + threadIdx.x;
    if (tid >= NNODES * F) return;
    const int f = tid % F;
    float v = in[tid] + bias[f];
    if (do_elu) v = v > 0.0f ? v : expm1f(v);
    out[tid] = v;
}

static inline int nblk(long long n, int b) { return (int)((n + b - 1) / b); }

extern "C" void kernel_launch(void* const* d_in, const int* in_sizes, int n_in,
                              void* d_out, int out_size, void* d_ws, size_t ws_size,
                              hipStream_t stream)
{
    (void)in_sizes; (void)n_in; (void)out_size; (void)ws_size;

    const float* x      = (const float*)d_in[0];
    const int*   ei     = (const int*)  d_in[1];   // [2, NEDGE]: row0=src, row1=dst
    const float* W1     = (const float*)d_in[2];
    const float* a1s    = (const float*)d_in[3];
    const float* a1d    = (const float*)d_in[4];
    const float* b1     = (const float*)d_in[5];
    const float* W2     = (const float*)d_in[6];
    const float* a2s    = (const float*)d_in[7];
    const float* a2d    = (const float*)d_in[8];
    const float* b2     = (const float*)d_in[9];
    const float* Wlin   = (const float*)d_in[10];
    const float* blin   = (const float*)d_in[11];

    // Output tuple: (out [N, NCLS], h [N, HC2]) concatenated flat.
    float* out_cls = (float*)d_out;                       // NNODES*NCLSV
    float* hout    = out_cls + (size_t)NNODES * NCLSV;    // NNODES*HC2

    // Workspace layout (floats)
    float* ws    = (float*)d_ws;
    float* t1    = ws;                                  // N*HC1   = 2,400,000
    float* t2    = t1   + (size_t)NNODES * HC1;         // N*HC2   = 5,120,000
    float* agg1  = t2   + (size_t)NNODES * HC2;         // N*HC1   (h1 after elu)
    float* asrc  = agg1 + (size_t)NNODES * HC1;         // N*HEADS = 80,000
    float* adst  = asrc + (size_t)NNODES * HEADS;
    unsigned* mu = (unsigned*)(adst + (size_t)NNODES * HEADS); // N*HEADS
    float* ssum  = (float*)mu + (size_t)NNODES * HEADS;        // N*HEADS
    float* alpha = ssum + (size_t)NNODES * HEADS;       // EEDGE*HEADS = 1,360,000

    const int eh_blocks = nblk((long long)EEDGE * HEADS, 256);

    // ============================= Layer 1 =============================
    hipMemsetAsync(agg1, 0, (size_t)NNODES * HC1 * sizeof(float), stream);
    hipMemsetAsync(mu,   0, (size_t)NNODES * HEADS * 2 * sizeof(float), stream); // mu + ssum

    // t1 = x @ W1 : [10000,512] @ [512,240]; NT=3 -> 48-wide strips, 240/48=5
    k_gemm_wmma_f32<3><<<dim3(NNODES / 16, HC1 / 48), 32, 0, stream>>>(
        x, W1, nullptr, t1, NNODES, FIN, HC1);

    k_attn_scores<<<nblk((long long)NNODES * HEADS, 256), 256, 0, stream>>>(
        t1, a1s, a1d, asrc, adst, C1V);

    k_edge_alpha_max<<<eh_blocks, 256, 0, stream>>>(ei, asrc, adst, alpha, mu);
    k_edge_expsum  <<<eh_blocks, 256, 0, stream>>>(ei, mu, alpha, ssum);
    k_edge_norm    <<<eh_blocks, 256, 0, stream>>>(ei, ssum, alpha);
    k_edge_aggregate<<<nblk((long long)EEDGE * HC1, 256), 256, 0, stream>>>(
        ei, alpha, t1, agg1, C1V);

    // h1 = elu(agg1 + b1), in place
    k_bias_act<<<nblk((long long)NNODES * HC1, 256), 256, 0, stream>>>(
        agg1, b1, agg1, HC1, /*do_elu=*/1);

    // ============================= Layer 2 =============================
    hipMemsetAsync(hout, 0, (size_t)NNODES * HC2 * sizeof(float), stream);
    hipMemsetAsync(mu,   0, (size_t)NNODES * HEADS * 2 * sizeof(float), stream);

    // t2 = h1 @ W2 : [10000,240] @ [240,512]; NT=4 -> 64-wide strips, 512/64=8
    k_gemm_wmma_f32<4><<<dim3(NNODES / 16, HC2 / 64), 32, 0, stream>>>(
        agg1, W2, nullptr, t2, NNODES, HC1, HC2);

    k_attn_scores<<<nblk((long long)NNODES * HEADS, 256), 256, 0, stream>>>(
        t2, a2s, a2d, asrc, adst, CHV);

    k_edge_alpha_max<<<eh_blocks, 256, 0, stream>>>(ei, asrc, adst, alpha, mu);
    k_edge_expsum  <<<eh_blocks, 256, 0, stream>>>(ei, mu, alpha, ssum);
    k_edge_norm    <<<eh_blocks, 256, 0, stream>>>(ei, ssum, alpha);
    k_edge_aggregate<<<nblk((long long)EEDGE * HC2, 256), 256, 0, stream>>>(
        ei, alpha, t2, hout, CHV);

    // h = agg2 + b2 (no activation) -> second output slice, in place
    k_bias_act<<<nblk((long long)NNODES * HC2, 256), 256, 0, stream>>>(
        hout, b2, hout, HC2, /*do_elu=*/0);

    // ======================== Classifier head ==========================
    // out = h @ Wlin + blin : [10000,512] @ [512,64]; NT=4 -> one 64-wide strip
    k_gemm_wmma_f32<4><<<dim3(NNODES / 16, NCLSV / 64), 32, 0, stream>>>(
        hout, Wlin, blin, out_cls, NNODES, HC2, NCLSV);
}